// SparseAttention_10599979286817
// MI455X (gfx1250) — compile-verified
//
#include <hip/hip_runtime.h>
#include <math.h>

typedef __attribute__((ext_vector_type(2))) float v2f;
typedef __attribute__((ext_vector_type(8))) float v8f;

#define S_DIM 2048
#define D_DIM 64
#define H_DIM 12
#define B_DIM 2
#define SSTR 2052   // padded score row stride (bank-conflict-free column reads)
#define QSTR 68     // padded Q / stage row stride
#define EPS  1e-6f

// Wait until this wave's outstanding async(global<->LDS) ops <= n
#define WAIT_ASYNC(n) asm volatile("s_wait_asynccnt %0" :: "i"(n) : "memory")

// Issue 8 async B128 loads per thread: 64x64 f32 tile (rows grow0..grow0+63 of a
// row-major [*,64] matrix) -> padded LDS buffer. Tracked on ASYNCcnt, no VGPR staging.
__device__ __forceinline__ void async_stage64x64(const float* __restrict__ gbase,
                                                 int grow0, float* lbuf, int tid)
{
#pragma unroll
    for (int j = 0; j < 8; ++j) {
        int g   = tid + j * 128;            // 16-byte group id, 1024 total
        int row = g >> 4;
        int c4  = (g & 15) * 4;             // stays inside a 64-float row: B128 never crosses pad
        unsigned lds_off = (unsigned)(size_t)(lbuf + row * QSTR + c4);   // low 32b = LDS offset
        unsigned goff    = (unsigned)(((grow0 + row) * D_DIM + c4) * 4);
        asm volatile("global_load_async_to_lds_b128 %0, %1, %2"
                     :: "v"(lds_off), "v"(goff), "s"(gbase) : "memory");
    }
}

__global__ __launch_bounds__(128) void sparse_attn_rms_kernel(
    const float* __restrict__ Q, const float* __restrict__ K,
    const float* __restrict__ V, const int* __restrict__ MSK,
    const float* __restrict__ W, float* __restrict__ Out,
    float* __restrict__ Pout)
{
    __shared__ float sls[16 * SSTR];        // full 16 x 2048 relu-score strip (padded)
    __shared__ float stage[2][64 * QSTR];   // double-buffered K/V chunk (async landing zone)
    __shared__ float qls[16 * QSTR];        // Q tile
    __shared__ float wls[S_DIM];            // rms_weight
    __shared__ float redls[4 * 32 * 8];     // sum-of-squares partials
    __shared__ float rinvls[16];            // 1/rms per row

    const int tid  = threadIdx.x;
    const int wave = tid >> 5;
    const int lane = tid & 31;
    const int nn   = lane & 15;             // N index within fragment
    const int mm   = lane & 15;             // M index for A fragment
    const int hi   = lane >> 4;             // lane-half select
    const int koff = hi * 2;                // K-pair base per ISA 16x4 f32 A/B layout

    const int q0 = blockIdx.x * 16;
    const int b  = blockIdx.z;
    const int bh = b * H_DIM + blockIdx.y;

    const float* Qp = Q + ((size_t)bh * S_DIM + q0) * D_DIM;
    const float* Kp = K + (size_t)bh * S_DIM * D_DIM;
    const float* Vp = V + (size_t)bh * S_DIM * D_DIM;
    const int*   Mp = MSK + ((size_t)b * S_DIM + q0) * S_DIM;

    // kick off K chunk 0 immediately (overlaps with Q/w staging below)
    async_stage64x64(Kp, 0, stage[0], tid);

    // ---- stage Q tile (16x64) and rms_weight, coalesced ----
#pragma unroll
    for (int j = 0; j < 8; ++j) {
        int e = tid + j * 128;
        qls[(e >> 6) * QSTR + (e & 63)] = Qp[e];
    }
    for (int c = tid; c < S_DIM; c += 128) wls[c] = W[c];

    float ss[8];
#pragma unroll
    for (int i = 0; i < 8; ++i) ss[i] = 0.f;

    // ---- Pass 1: scores = relu(mask(QK^T/8)) into LDS, accumulate sum(s^2) ----
    int cur = 0;
    for (int ktg = 0; ktg < S_DIM; ktg += 64) {
        if (ktg + 64 < S_DIM) {             // software pipeline: next chunk in flight
            async_stage64x64(Kp, ktg + 64, stage[cur ^ 1], tid);
            WAIT_ASYNC(8);                  // my current-chunk loads have landed
        } else {
            WAIT_ASYNC(0);
        }
        __syncthreads();                    // => every wave's loads have landed

        const float* st = stage[cur];
        const int kt = ktg + wave * 16;
        v8f c0 = {}, c1 = {};               // two chains: double XDL ILP
#pragma unroll
        for (int kk = 0; kk < 32; kk += 4) {
            v2f a, bv;
            a.x  = qls[mm * QSTR + kk + koff];
            a.y  = qls[mm * QSTR + kk + koff + 1];
            bv.x = st[(wave * 16 + nn) * QSTR + kk + koff];
            bv.y = st[(wave * 16 + nn) * QSTR + kk + koff + 1];
            c0 = __builtin_amdgcn_wmma_f32_16x16x4_f32(false, a, false, bv,
                                                       (short)0, c0, false, false);
            a.x  = qls[mm * QSTR + kk + 32 + koff];
            a.y  = qls[mm * QSTR + kk + 32 + koff + 1];
            bv.x = st[(wave * 16 + nn) * QSTR + kk + 32 + koff];
            bv.y = st[(wave * 16 + nn) * QSTR + kk + 32 + koff + 1];
            c1 = __builtin_amdgcn_wmma_f32_16x16x4_f32(false, a, false, bv,
                                                       (short)0, c1, false, false);
        }
#pragma unroll
        for (int i = 0; i < 8; ++i) {
            int r = i + 8 * hi;                       // C/D layout: VGPR i -> row i (+8 hi half)
            float s = (c0[i] + c1[i]) * 0.125f;       // 1/sqrt(64)
            int mk = Mp[(size_t)r * S_DIM + kt + nn];
            s = (mk != 0 && s > 0.f) ? s : 0.f;       // mask->NEG then relu == 0
            ss[i] += s * s;
            sls[r * SSTR + kt + nn] = s;
        }
        __syncthreads();                    // protect stage[cur] before it is re-targeted
        cur ^= 1;
    }

    // prefetch V chunk 0 now: latency hides under reduction + p_attn streamout
    async_stage64x64(Vp, 0, stage[0], tid);

    // ---- deterministic sum-of-squares reduction -> 1/rms ----
#pragma unroll
    for (int i = 0; i < 8; ++i) redls[(wave * 32 + lane) * 8 + i] = ss[i];
    __syncthreads();
    if (tid < 16) {
        int rhi = tid >> 3, ri = tid & 7;
        float t = 0.f;
        for (int w2 = 0; w2 < 4; ++w2)
            for (int l = 16 * rhi; l < 16 * rhi + 16; ++l)
                t += redls[(w2 * 32 + l) * 8 + ri];
        rinvls[tid] = 1.0f / sqrtf(t * (1.0f / S_DIM) + EPS);
    }
    __syncthreads();

    // ---- Pass 2a: fold w[k] into LDS scores, stream normalized p_attn out ----
    float* prow0 = Pout + ((size_t)bh * S_DIM + q0) * S_DIM;
    for (int r = 0; r < 16; ++r) {
        float ri = rinvls[r];
        float* prow = prow0 + (size_t)r * S_DIM;
        float* srow = sls + r * SSTR;
        for (int cc = tid; cc < S_DIM; cc += 128) {
            float s = srow[cc] * wls[cc];
            srow[cc] = s;
            prow[cc] = s * ri;                        // coalesced streamout (dominant traffic)
        }
    }
    __syncthreads();

    // ---- Pass 2b: out = (P .* w) @ V, scaled by 1/rms at the end ----
    const int nw = wave * 16;
    v8f co0 = {}, co1 = {};
    cur = 0;
    for (int kkc = 0; kkc < S_DIM; kkc += 64) {
        if (kkc + 64 < S_DIM) {
            async_stage64x64(Vp, kkc + 64, stage[cur ^ 1], tid);
            WAIT_ASYNC(8);
        } else {
            WAIT_ASYNC(0);
        }
        __syncthreads();

        const float* st = stage[cur];
#pragma unroll
        for (int kk = 0; kk < 32; kk += 4) {
            v2f a, bv;
            a.x  = sls[mm * SSTR + kkc + kk + koff];
            a.y  = sls[mm * SSTR + kkc + kk + koff + 1];
            bv.x = st[(kk + koff) * QSTR + nw + nn];
            bv.y = st[(kk + koff + 1) * QSTR + nw + nn];
            co0 = __builtin_amdgcn_wmma_f32_16x16x4_f32(false, a, false, bv,
                                                        (short)0, co0, false, false);
            a.x  = sls[mm * SSTR + kkc + kk + 32 + koff];
            a.y  = sls[mm * SSTR + kkc + kk + 32 + koff + 1];
            bv.x = st[(kk + 32 + koff) * QSTR + nw + nn];
            bv.y = st[(kk + 32 + koff + 1) * QSTR + nw + nn];
            co1 = __builtin_amdgcn_wmma_f32_16x16x4_f32(false, a, false, bv,
                                                        (short)0, co1, false, false);
        }
        __syncthreads();
        cur ^= 1;
    }
#pragma unroll
    for (int i = 0; i < 8; ++i) {
        int r = i + 8 * hi;
        Out[((size_t)bh * S_DIM + q0 + r) * D_DIM + nw + nn] = (co0[i] + co1[i]) * rinvls[r];
    }
}

extern "C" void kernel_launch(void* const* d_in, const int* in_sizes, int n_in,
                              void* d_out, int out_size, void* d_ws, size_t ws_size,
                              hipStream_t stream) {
    (void)in_sizes; (void)n_in; (void)d_ws; (void)ws_size; (void)out_size;
    const float* Q   = (const float*)d_in[0];
    const float* K   = (const float*)d_in[1];
    const float* V   = (const float*)d_in[2];
    const int*   MSK = (const int*)d_in[3];
    const float* W   = (const float*)d_in[4];

    float* Out  = (float*)d_out;                                   // [B,H,S,D]
    float* Pout = Out + (size_t)B_DIM * H_DIM * S_DIM * D_DIM;     // [B,H,S,S]

    dim3 grid(S_DIM / 16, H_DIM, B_DIM);
    dim3 block(128);
    sparse_attn_rms_kernel<<<grid, block, 0, stream>>>(Q, K, V, MSK, W, Out, Pout);
}